// MoEFusion_24068996727394
// MI455X (gfx1250) — compile-verified
//
#include <hip/hip_runtime.h>
#include <hip/hip_bf16.h>
#include <math.h>

typedef __attribute__((ext_vector_type(2))) float v2f;
typedef __attribute__((ext_vector_type(4))) float v4f;
typedef __attribute__((ext_vector_type(8))) float v8f;

// ---------------------------------------------------------------------------
// Kernel 1: router logits via V_WMMA_F32_16X16X4_F32, then per-lane top-2
// softmax scatter. One wave (32 threads) handles a tile of 16 tokens.
//
// logits^T (M=expert, N=token) = W(16pad x K) @ hidden^T(K x 16 tokens):
// with the 16x16 f32 C/D layout, lane t (t<16) holds all 8 expert logits for
// token t in its 8 accumulator VGPRs -> per-lane scalar top-2, no cross-lane.
//
// Per iteration: batch 8 A-fragments + 8 B-fragments (K=32) into registers
// first (one load clause, latency amortized), then 8 WMMAs round-robin over
// 4 independent accumulator chains to keep the matrix pipe fed.
// ---------------------------------------------------------------------------
__global__ __launch_bounds__(32)
void moe_router_topk_kernel(const float* __restrict__ hidden,
                            const float* __restrict__ rw,
                            float* __restrict__ weights,
                            int D, int nTokens) {
    const int lane = threadIdx.x;
    const int l    = lane & 15;   // row/col within tile
    const int kh   = lane >> 4;   // K-half: 0 -> K={0,1}, 1 -> K={2,3}
    const int t0   = blockIdx.x * 16;

    const int erow = l & 7;                       // pad experts 8..15 as dups (rows unused)
    int tok = t0 + l; if (tok >= nTokens) tok = nTokens - 1;

    const float* aP = rw     + (size_t)erow * D + kh * 2;
    const float* bP = hidden + (size_t)tok  * D + kh * 2;

    v8f acc0 = {}, acc1 = {}, acc2 = {}, acc3 = {};

    for (int k = 0; k < D; k += 32) {           // 8 WMMAs (K=32) per iteration
        v2f a[8], b[8];
#pragma unroll
        for (int j = 0; j < 8; ++j) {
            a[j] = *(const v2f*)(aP + k + 4 * j);                              // W: cached
            b[j] = __builtin_nontemporal_load((const v2f*)(bP + k + 4 * j));   // hidden: NT
        }
        acc0 = __builtin_amdgcn_wmma_f32_16x16x4_f32(false, a[0], false, b[0], (short)0, acc0, false, false);
        acc1 = __builtin_amdgcn_wmma_f32_16x16x4_f32(false, a[1], false, b[1], (short)0, acc1, false, false);
        acc2 = __builtin_amdgcn_wmma_f32_16x16x4_f32(false, a[2], false, b[2], (short)0, acc2, false, false);
        acc3 = __builtin_amdgcn_wmma_f32_16x16x4_f32(false, a[3], false, b[3], (short)0, acc3, false, false);
        acc0 = __builtin_amdgcn_wmma_f32_16x16x4_f32(false, a[4], false, b[4], (short)0, acc0, false, false);
        acc1 = __builtin_amdgcn_wmma_f32_16x16x4_f32(false, a[5], false, b[5], (short)0, acc1, false, false);
        acc2 = __builtin_amdgcn_wmma_f32_16x16x4_f32(false, a[6], false, b[6], (short)0, acc2, false, false);
        acc3 = __builtin_amdgcn_wmma_f32_16x16x4_f32(false, a[7], false, b[7], (short)0, acc3, false, false);
    }
    v8f c = (acc0 + acc1) + (acc2 + acc3);

    // lanes 0..15 each own one token's 8 expert logits (vgprs 0..7 = experts 0..7)
    if (kh == 0 && (t0 + l) < nTokens) {
        float lg[8];
#pragma unroll
        for (int n = 0; n < 8; ++n) lg[n] = c[n];

        // top-1 (ties -> lowest index, matches jax.lax.top_k)
        int bi = 0; float bv = lg[0];
#pragma unroll
        for (int n = 1; n < 8; ++n) if (lg[n] > bv) { bv = lg[n]; bi = n; }
        // top-2
        int si = -1; float sv = -INFINITY;
#pragma unroll
        for (int n = 0; n < 8; ++n) if (n != bi && lg[n] > sv) { sv = lg[n]; si = n; }

        // softmax over the two kept logits
        float e  = __expf(sv - bv);
        float wb = 1.0f / (1.0f + e);
        float ws = e * wb;

        float wout[8];
#pragma unroll
        for (int n = 0; n < 8; ++n) wout[n] = (n == bi) ? wb : ((n == si) ? ws : 0.0f);

        v4f o0 = { wout[0], wout[1], wout[2], wout[3] };
        v4f o1 = { wout[4], wout[5], wout[6], wout[7] };
        float* wp = weights + (size_t)(t0 + l) * 8;
        *(v4f*)(wp)     = o0;
        *(v4f*)(wp + 4) = o1;
    }
}

// ---------------------------------------------------------------------------
// Kernel 2: bandwidth-bound fusion (536 MB of the ~670 MB total traffic).
// expert_outputs is (tok, d, n=8) with the 8 experts contiguous -> two B128
// NT loads per output element, dot with the token's 8 weights in registers,
// NT store. One 256-thread block per token; lane-adjacent d => perfectly
// coalesced 32 B/lane streams. Unroll to keep many B128 loads in flight.
// ---------------------------------------------------------------------------
__global__ __launch_bounds__(256)
void moe_fuse_kernel(const float* __restrict__ eo,
                     const float* __restrict__ weights,
                     float* __restrict__ fused,
                     int D) {
    const long long token = blockIdx.x;
    const int tid = threadIdx.x;

    const v4f* wp = (const v4f*)(weights + token * 8);
    const v4f w0 = wp[0];
    const v4f w1 = wp[1];

    const float* base = eo    + token * (long long)D * 8;
    float*       outp = fused + token * (long long)D;

#pragma unroll 4
    for (int d = tid; d < D; d += 256) {
        const v4f* p = (const v4f*)(base + (size_t)d * 8);
        v4f e0 = __builtin_nontemporal_load(p);
        v4f e1 = __builtin_nontemporal_load(p + 1);
        float acc = e0.x * w0.x + e0.y * w0.y + e0.z * w0.z + e0.w * w0.w;
        acc      += e1.x * w1.x + e1.y * w1.y + e1.z * w1.z + e1.w * w1.w;
        __builtin_nontemporal_store(acc, outp + d);
    }
}

extern "C" void kernel_launch(void* const* d_in, const int* in_sizes, int n_in,
                              void* d_out, int out_size, void* d_ws, size_t ws_size,
                              hipStream_t stream) {
    const float* eo     = (const float*)d_in[0];   // (B,T,D,N) f32
    const float* hidden = (const float*)d_in[1];   // (B,T,D)   f32
    const float* rw     = (const float*)d_in[2];   // (N,D)     f32

    const long long eoN = in_sizes[0];
    const long long hN  = in_sizes[1];
    const long long rN  = in_sizes[2];
    const int N       = (int)(eoN / hN);   // 8
    const int D       = (int)(rN / N);     // 2048
    const int nTokens = (int)(hN / D);     // 8192
    (void)N;

    float* fused   = (float*)d_out;                       // (B,T,D)
    float* weights = fused + (size_t)nTokens * D;         // (B,T,N) tail of d_out

    const int nTiles = (nTokens + 15) / 16;
    moe_router_topk_kernel<<<nTiles, 32, 0, stream>>>(hidden, rw, weights, D, nTokens);
    moe_fuse_kernel<<<nTokens, 256, 0, stream>>>(eo, weights, fused, D);
}